// Huffmax_15083925143710
// MI455X (gfx1250) — compile-verified
//
#include <hip/hip_runtime.h>
#include <hip/hip_bf16.h>

// Huffmax hierarchical softmax for MI455X (gfx1250, wave32).
//
// Phase 1: dense f32 GEMM via V_WMMA_F32_16X16X4_F32 computing
//          Ytab[b][j] = sigmoid(X[b] . W[thresh+j] + bias[thresh+j])
//          for the NTOP highest node ids (top-of-Huffman-tree nodes that
//          appear in ~11 of 16 path entries of every request).
// Phase 2: one wave per (batch, request): walk the D path entries; table hit
//          for high node ids, cooperative wave32 dot-512 + shfl_xor butterfly
//          reduction for deep (rare, per-request-unique) nodes.

typedef float v2f __attribute__((ext_vector_type(2)));
typedef float v8f __attribute__((ext_vector_type(8)));

#define IDIM 512   // input_dim, fixed by reference

// ---------------------------------------------------------------------------
// Phase 1: top-node GEMM.  One wave computes a 16(M) x 64(N) output tile.
// A = X (M=batch rows), B = W_top^T (columns are W rows), K = 512.
// f32 WMMA fragment layouts (ISA 7.12.2):
//   A 16x4:  lane<16: {A[m=lane][k0], A[m][k0+1]}, lane>=16: {A[m][k0+2], A[m][k0+3]}
//   B 4x16:  lane<16: {B[k0][n=lane], B[k0+1][n]}, lane>=16: {B[k0+2][n], B[k0+3][n]}
//   D 16x16: vgpr g: lane<16 -> (m0+g, n=lane), lane>=16 -> (m0+8+g, n=lane-16)
// ---------------------------------------------------------------------------
__global__ void __launch_bounds__(256)
huffmax_top_gemm(const float* __restrict__ X, const float* __restrict__ W,
                 const float* __restrict__ bias, float* __restrict__ Ytab,
                 int mtiles, int nt64, int ntop, int nodeBase)
{
    const int lane = threadIdx.x & 31;
    const int wave = (int)((blockIdx.x * blockDim.x + threadIdx.x) >> 5);
    const int mt = wave / nt64;
    const int nt = wave - mt * nt64;
    if (mt >= mtiles) return;

    const int m0   = mt << 4;   // 16 batch rows
    const int n0   = nt << 6;   // 64 node columns
    const int half = lane >> 4; // selects K pair within the 4-wide K step
    const int l15  = lane & 15;

    // per-lane base pointers (contiguous 8-byte loads along K)
    const float* ap = X + (size_t)(m0 + l15) * IDIM + half * 2;
    const float* bp = W + (size_t)(nodeBase + n0 + l15) * IDIM + half * 2;

    v8f acc0 = {}, acc1 = {}, acc2 = {}, acc3 = {};

    for (int k = 0; k < IDIM; k += 4) {
        v2f a  = *(const v2f*)(ap + k);
        v2f b0 = *(const v2f*)(bp + k);
        v2f b1 = *(const v2f*)(bp + 16 * IDIM + k);
        v2f b2 = *(const v2f*)(bp + 32 * IDIM + k);
        v2f b3 = *(const v2f*)(bp + 48 * IDIM + k);
#if __has_builtin(__builtin_amdgcn_wmma_f32_16x16x4_f32)
        acc0 = __builtin_amdgcn_wmma_f32_16x16x4_f32(false, a, false, b0, (short)0, acc0, false, false);
        acc1 = __builtin_amdgcn_wmma_f32_16x16x4_f32(false, a, false, b1, (short)0, acc1, false, false);
        acc2 = __builtin_amdgcn_wmma_f32_16x16x4_f32(false, a, false, b2, (short)0, acc2, false, false);
        acc3 = __builtin_amdgcn_wmma_f32_16x16x4_f32(false, a, false, b3, (short)0, acc3, false, false);
#else
        // VALU fallback (keeps compile green if the f32 WMMA builtin name differs;
        // mimics the WMMA D fragment layout so the epilogue stays identical).
        #pragma unroll
        for (int g = 0; g < 8; ++g) {
            // D element (m0 + g + 8*half, n0 + 16j + l15): each lane owns rows
            // m = g + 8*half of the A tile; recompute A values via shuffle-free
            // re-load (cheap: L2 hit).
            const float* am = X + (size_t)(m0 + g + 8 * half) * IDIM + k;
            float a0 = am[0], a1 = am[1], a2 = am[2], a3 = am[3];
            const float* wn = W + (size_t)(nodeBase + n0 + l15) * IDIM + k;
            acc0[g] += a0 * wn[0] + a1 * wn[1] + a2 * wn[2] + a3 * wn[3];
            wn += 16 * IDIM;
            acc1[g] += a0 * wn[0] + a1 * wn[1] + a2 * wn[2] + a3 * wn[3];
            wn += 16 * IDIM;
            acc2[g] += a0 * wn[0] + a1 * wn[1] + a2 * wn[2] + a3 * wn[3];
            wn += 16 * IDIM;
            acc3[g] += a0 * wn[0] + a1 * wn[1] + a2 * wn[2] + a3 * wn[3];
        }
#endif
    }

    // Epilogue: bias + sigmoid, scatter per D-fragment layout.
    const int mrow = m0 + half * 8;
    #pragma unroll
    for (int j = 0; j < 4; ++j) {
        v8f acc = (j == 0) ? acc0 : (j == 1) ? acc1 : (j == 2) ? acc2 : acc3;
        const int n = n0 + j * 16 + l15;
        const float bv = bias[nodeBase + n];
        #pragma unroll
        for (int g = 0; g < 8; ++g) {
            float v = acc[g] + bv;
            Ytab[(size_t)(mrow + g) * ntop + n] = 1.0f / (1.0f + __expf(-v));
        }
    }
}

// ---------------------------------------------------------------------------
// Phase 2: one wave32 per (batch, request).
// ---------------------------------------------------------------------------
__global__ void __launch_bounds__(256)
huffmax_eval(const float* __restrict__ X, const int* __restrict__ tgt,
             const float* __restrict__ W, const float* __restrict__ bias,
             const int* __restrict__ paths, const float* __restrict__ codes,
             const float* __restrict__ Ytab, float* __restrict__ out,
             int nwaves, int R, int D, int ntop, int thresh)
{
    const int lane = threadIdx.x & 31;
    const int wid  = (int)((blockIdx.x * blockDim.x + threadIdx.x) >> 5);
    if (wid >= nwaves) return;

    const int b = wid / R;
    const int r = wid - b * R;
    const int cls = tgt[b * R + r];

    // preload x_b: 16 floats per lane, lane-interleaved for coalescing
    float xr[16];
    const float* xp = X + (size_t)b * IDIM;
    #pragma unroll
    for (int i = 0; i < 16; ++i) xr[i] = xp[i * 32 + lane];

    float prod = 1.0f;
    const int*   pp = paths + (size_t)cls * D;
    const float* cp = codes + (size_t)cls * D;

    for (int d = 0; d < D; ++d) {
        const int   node = pp[d];           // wave-uniform
        const float code = cp[d];
        float y;
        if (node >= thresh) {
            y = Ytab[(size_t)b * ntop + (node - thresh)];   // broadcast hit
        } else {
            const float* wp = W + (size_t)node * IDIM;
            float s = 0.0f;
            #pragma unroll
            for (int i = 0; i < 16; ++i) s = fmaf(xr[i], wp[i * 32 + lane], s);
            // wave32 butterfly: all lanes end with the full sum
            #pragma unroll
            for (int off = 16; off >= 1; off >>= 1) s += __shfl_xor(s, off, 32);
            y = 1.0f / (1.0f + __expf(-(s + bias[node])));
        }
        prod *= (code + y - 2.0f * code * y);
    }

    if (lane == 0) out[wid] = prod;
}

extern "C" void kernel_launch(void* const* d_in, const int* in_sizes, int n_in,
                              void* d_out, int out_size, void* d_ws, size_t ws_size,
                              hipStream_t stream)
{
    const float* X     = (const float*)d_in[0];   // (B, 512)
    const int*   tgt   = (const int*)  d_in[1];   // (B, R)
    const float* W     = (const float*)d_in[2];   // (total_nodes, 512)
    const float* bias  = (const float*)d_in[3];   // (total_nodes,)
    const int*   paths = (const int*)  d_in[4];   // (nb_classes, D)
    const float* codes = (const float*)d_in[5];   // (nb_classes, D)
    float*       out   = (float*)d_out;

    const int Bsz        = in_sizes[0] / IDIM;          // 1024
    const int R          = in_sizes[1] / Bsz;           // 8
    const int totalNodes = in_sizes[2] / IDIM;          // 50256
    const int nclasses   = totalNodes + 1;              // 50257
    const int D          = in_sizes[4] / nclasses;      // 16

    // Size the top-node table by available workspace (cap 2048 ids = 8 MB f32).
    int ntop = 0;
    if (ws_size >= (size_t)Bsz * 64 * sizeof(float)) {
        ntop = (int)(ws_size / ((size_t)Bsz * sizeof(float)));
        ntop = (ntop / 64) * 64;
        if (ntop > 2048) ntop = 2048;
        if (ntop > totalNodes) ntop = (totalNodes / 64) * 64;
    }
    const int thresh = totalNodes - ntop;               // ids >= thresh hit the table

    if (ntop > 0) {
        const int mtiles = Bsz / 16;
        const int nt64   = ntop / 64;
        const int waves  = mtiles * nt64;
        const int block  = 256;
        const int grid   = (waves * 32 + block - 1) / block;
        huffmax_top_gemm<<<grid, block, 0, stream>>>(X, W, bias, (float*)d_ws,
                                                     mtiles, nt64, ntop, thresh);
    }

    const int nwaves = Bsz * R;                         // 8192
    const int block2 = 256;                             // 8 waves / block
    const int grid2  = (nwaves * 32 + block2 - 1) / block2;
    huffmax_eval<<<grid2, block2, 0, stream>>>(X, tgt, W, bias, paths, codes,
                                               (const float*)d_ws, out,
                                               nwaves, R, D, ntop, thresh);
}